// ClinicalROILoss_37263136260896
// MI455X (gfx1250) — compile-verified
//
#include <hip/hip_runtime.h>
#include <hip/hip_bf16.h>

// ---------------------------------------------------------------------------
// ClinicalROILoss for MI455X (gfx1250, wave32).
// Pipeline: surface/erosion -> 3x in-place min-plus EDT passes (squared dist)
//           -> fused 25-sum reduction (WMMA f32 16x16x4 wave reduction)
//           -> deterministic partial combine -> on-device radix-select p95
//           -> scalar finalize. No host sync, graph-capture safe,
//           bit-deterministic (integer LDS atomics only, fixed-order sums).
// ---------------------------------------------------------------------------

static constexpr int DD = 64, HH = 96, WW = 96;
static constexpr int HW = HH * WW;            // 9216
static constexpr int NN = DD * HH * WW;       // 589824
static constexpr int NB = 288;                // reduction blocks
static constexpr int NSUMS = 25;
#define EDT_INF 1.0e12f

typedef __attribute__((ext_vector_type(2))) float v2f;
typedef __attribute__((ext_vector_type(8))) float v8f;

// Sum v across the 32 lanes of a wave. WMMA path: A(16x4) vgpr0 = v (K=0 for
// lanes 0-15, K=2 for lanes 16-31), vgpr1 = 0; B = ones(4x16)
// => D[m][n] = v[m] + v[m+16]. Sum D's 8 VGPRs per lane, then lane0 + lane16.
__device__ __forceinline__ float wave_reduce_sum(float v) {
#if defined(__AMDGCN__) && __has_builtin(__builtin_amdgcn_wmma_f32_16x16x4_f32)
  v2f a; a.x = v; a.y = 0.0f;
  v2f b; b.x = 1.0f; b.y = 1.0f;
  v8f c = {};
  c = __builtin_amdgcn_wmma_f32_16x16x4_f32(false, a, false, b, (short)0, c,
                                            false, false);
  float s = c[0] + c[1] + c[2] + c[3] + c[4] + c[5] + c[6] + c[7];
  int si = __float_as_int(s);
  float t0 = __int_as_float(__builtin_amdgcn_readlane(si, 0));
  float t1 = __int_as_float(__builtin_amdgcn_readlane(si, 16));
  return t0 + t1;
#else
  for (int o = 16; o > 0; o >>= 1) v += __shfl_xor(v, o, 32);
  return v;
#endif
}

// --- K1: binary surface (mask - erosion) + EDT field init -------------------
__global__ __launch_bounds__(256) void k_surface(
    const float* __restrict__ lp, const float* __restrict__ lg,
    float* __restrict__ fp, float* __restrict__ ft,
    unsigned char* __restrict__ sp, unsigned char* __restrict__ st) {
  int idx = blockIdx.x * blockDim.x + threadIdx.x;
  if (idx >= NN) return;
  int w = idx % WW;
  int h = (idx / WW) % HH;
  int d = idx / HW;
  auto bp = [&](int dd, int hh, int ww) -> bool {
    if (dd < 0 || dd >= DD || hh < 0 || hh >= HH || ww < 0 || ww >= WW)
      return false;
    return lp[(dd * HH + hh) * WW + ww] > 0.5f;
  };
  auto bg = [&](int dd, int hh, int ww) -> bool {
    if (dd < 0 || dd >= DD || hh < 0 || hh >= HH || ww < 0 || ww >= WW)
      return false;
    return lg[(dd * HH + hh) * WW + ww] > 0.5f;
  };
  {
    bool c = lp[idx] > 0.5f;
    bool er = c && bp(d - 1, h, w) && bp(d + 1, h, w) && bp(d, h - 1, w) &&
              bp(d, h + 1, w) && bp(d, h, w - 1) && bp(d, h, w + 1);
    bool s = c && !er;
    sp[idx] = s ? 1 : 0;
    fp[idx] = s ? 0.0f : EDT_INF;
  }
  {
    bool c = lg[idx] > 0.5f;
    bool er = c && bg(d - 1, h, w) && bg(d + 1, h, w) && bg(d, h - 1, w) &&
              bg(d, h + 1, w) && bg(d, h, w - 1) && bg(d, h, w + 1);
    bool s = c && !er;
    st[idx] = s ? 1 : 0;
    ft[idx] = s ? 0.0f : EDT_INF;
  }
}

// --- K2: one separable squared-EDT pass along `axis`, in place --------------
// grid.x = #lines, grid.y = field (0: pred, 1: targ), block = 96 threads.
__global__ __launch_bounds__(96) void k_edt(float* __restrict__ fpred,
                                            float* __restrict__ ftarg,
                                            int axis) {
  __shared__ float ls[96];
  float* f = (blockIdx.y == 0) ? fpred : ftarg;
  int line = blockIdx.x;
  int len, stride, base;
  if (axis == 0) { len = DD; stride = HW; base = line; }
  else if (axis == 1) { len = HH; stride = WW; base = (line / WW) * HW + (line % WW); }
  else { len = WW; stride = 1; base = line * WW; }
  int i = threadIdx.x;
  if (i < len) ls[i] = f[base + i * stride];
  __syncthreads();
  if (i < len) {
    float best = 3.4e38f;
    #pragma unroll 4
    for (int j = 0; j < len; ++j) {
      float dt = (float)(i - j);
      best = fminf(best, fmaf(dt, dt, ls[j]));   // spacing = 1
    }
    f[base + i * stride] = best;
  }
}

// --- K3: fused 25-way reduction over the volume -----------------------------
// sums 0..8  : brain SSIM  {Sm, Spm, Stm, Spm2, Stm2, Sp2m2, St2m2, Sptm2, Sm2}
// sums 9..17 : bone  SSIM  (same, t = ct, m = bone)
// sums 18..20: dice {S lp*lg, S lp, S lg}
// 21: count(ts & fp<=4)  22: count(ps & ft<=4)  23: ps_n  24: ts_n
__global__ __launch_bounds__(256) void k_reduce(
    const float* __restrict__ fused, const float* __restrict__ mri,
    const float* __restrict__ ct, const float* __restrict__ brain,
    const float* __restrict__ bone, const float* __restrict__ lp,
    const float* __restrict__ lg, const float* __restrict__ fp,
    const float* __restrict__ ft, const unsigned char* __restrict__ sp,
    const unsigned char* __restrict__ st, float* __restrict__ partials) {
  float acc[NSUMS];
  #pragma unroll
  for (int s = 0; s < NSUMS; ++s) acc[s] = 0.0f;
  int tid = threadIdx.x;
  int gsz = gridDim.x * blockDim.x;
  for (int i = blockIdx.x * blockDim.x + tid; i < NN; i += gsz) {
    float p = fused[i], t1 = mri[i], t2 = ct[i];
    float m1 = brain[i], m2 = bone[i];
    float a = lp[i], g = lg[i];
    float dp = fp[i], dt2 = ft[i];
    bool bsp = sp[i] != 0, bst = st[i] != 0;
    float m1s = m1 * m1, m2s = m2 * m2;
    acc[0] += m1;        acc[1] += p * m1;     acc[2] += t1 * m1;
    acc[3] += p * m1s;   acc[4] += t1 * m1s;
    acc[5] += p * p * m1s; acc[6] += t1 * t1 * m1s;
    acc[7] += p * t1 * m1s; acc[8] += m1s;
    acc[9] += m2;        acc[10] += p * m2;    acc[11] += t2 * m2;
    acc[12] += p * m2s;  acc[13] += t2 * m2s;
    acc[14] += p * p * m2s; acc[15] += t2 * t2 * m2s;
    acc[16] += p * t2 * m2s; acc[17] += m2s;
    acc[18] += a * g;    acc[19] += a;         acc[20] += g;
    acc[21] += (bst && dp <= 4.0f) ? 1.0f : 0.0f;   // tol^2 = 4
    acc[22] += (bsp && dt2 <= 4.0f) ? 1.0f : 0.0f;
    acc[23] += bsp ? 1.0f : 0.0f;
    acc[24] += bst ? 1.0f : 0.0f;
  }
  __shared__ float lds[NSUMS][8];
  int wave = tid >> 5, lane = tid & 31;
  #pragma unroll
  for (int s = 0; s < NSUMS; ++s) {
    float r = wave_reduce_sum(acc[s]);   // v_wmma_f32_16x16x4_f32
    if (lane == 0) lds[s][wave] = r;
  }
  __syncthreads();
  if (tid < NSUMS) {
    float t = 0.0f;
    #pragma unroll
    for (int w2 = 0; w2 < 8; ++w2) t += lds[tid][w2];
    partials[tid * gridDim.x + blockIdx.x] = t;
  }
}

// --- K4: combine per-block partials in fixed order (deterministic) ----------
__global__ void k_finalize(const float* __restrict__ partials,
                           float* __restrict__ scalars) {
  int tid = threadIdx.x;
  if (tid < NSUMS) {
    float t = 0.0f;
    for (int b = 0; b < NB; ++b) t += partials[tid * NB + b];
    scalars[tid] = t;
  }
}

// --- K5: block-cooperative radix select (k-th smallest masked value) --------
__device__ float radix_select(const float* __restrict__ field,
                              const unsigned char* __restrict__ mask, int k,
                              unsigned int* hist, unsigned int* sh_prefix,
                              int* sh_k, int tid) {
  unsigned int prefix = 0;
  for (int round = 0; round < 4; ++round) {
    int shift = 24 - 8 * round;
    hist[tid] = 0;            // blockDim == 256
    __syncthreads();
    for (int i = tid; i < NN; i += 256) {
      if (mask[i]) {
        unsigned int bits = __float_as_uint(field[i]);  // all values >= 0
        bool match = (round == 0) || ((bits >> (shift + 8)) == prefix);
        if (match) atomicAdd(&hist[(bits >> shift) & 0xFFu], 1u);
      }
    }
    __syncthreads();
    if (tid == 0) {
      unsigned int cum = 0; int bin = 255; int kk = k;
      for (int b = 0; b < 256; ++b) {
        unsigned int c = hist[b];
        if (cum + c > (unsigned int)k) { bin = b; kk = k - (int)cum; break; }
        cum += c;
      }
      *sh_prefix = (prefix << 8) | (unsigned int)bin;
      *sh_k = kk;
    }
    __syncthreads();
    prefix = *sh_prefix;
    k = *sh_k;
    __syncthreads();
  }
  return __uint_as_float(prefix);
}

__global__ __launch_bounds__(256) void k_select(
    const float* __restrict__ fpred, const float* __restrict__ ftarg,
    const unsigned char* __restrict__ sp, const unsigned char* __restrict__ st,
    float* __restrict__ scalars) {
  __shared__ unsigned int hist[256];
  __shared__ unsigned int sh_prefix;
  __shared__ int sh_k;
  const float* field; const unsigned char* mask; float nf;
  if (blockIdx.x == 0) { field = fpred; mask = st; nf = scalars[24]; }
  else                 { field = ftarg; mask = sp; nf = scalars[23]; }
  int tid = threadIdx.x;
  int n = (int)(nf + 0.5f);
  float result = 0.0f;
  if (n > 0) {
    float pos = 0.95f * (float)(n - 1);
    int lo = (int)floorf(pos);
    int hi = (int)ceilf(pos);
    float wgt = pos - (float)lo;
    float vlo = radix_select(field, mask, lo, hist, &sh_prefix, &sh_k, tid);
    float vhi = (hi == lo)
                    ? vlo
                    : radix_select(field, mask, hi, hist, &sh_prefix, &sh_k, tid);
    result = sqrtf(vlo) * (1.0f - wgt) + sqrtf(vhi) * wgt;  // sqrt monotone
  }
  if (tid == 0) scalars[25 + blockIdx.x] = result;
}

// --- K6: scalar finalize ----------------------------------------------------
__global__ void k_final(const float* __restrict__ sc, float* __restrict__ out) {
  auto ssim = [&](int o) -> float {
    float Sm = sc[o + 0], Spm = sc[o + 1], Stm = sc[o + 2];
    float Spm2 = sc[o + 3], Stm2 = sc[o + 4];
    float Sp2m2 = sc[o + 5], St2m2 = sc[o + 6], Sptm2 = sc[o + 7];
    float Sm2 = sc[o + 8];
    float n = Sm + 1e-8f;
    float mup = Spm / n, mut = Stm / n;
    float sigp = (Sp2m2 - 2.0f * mup * Spm2 + mup * mup * Sm2) / n;
    float sigt = (St2m2 - 2.0f * mut * Stm2 + mut * mut * Sm2) / n;
    float sigpt = (Sptm2 - mup * Stm2 - mut * Spm2 + mup * mut * Sm2) / n;
    const float C1 = 1e-4f, C2 = 9e-4f;
    float num = (2.0f * mup * mut + C1) * (2.0f * sigpt + C2);
    float den = (mup * mup + mut * mut + C1) * (sigp + sigt + C2);
    float s = num / (den + 1e-8f);
    return fminf(fmaxf(s, 0.0f), 1.0f);
  };
  float l_brain = 1.0f - ssim(0);
  float l_bone = 1.0f - ssim(9);
  float l_dice = 1.0f - (2.0f * sc[18] + 1.0f) / (sc[19] + sc[20] + 1.0f);
  float psn = sc[23], tsn = sc[24];
  bool empty = (psn < 0.5f) || (tsn < 0.5f);
  float hd95 = empty ? 100.0f : fmaxf(sc[25], sc[26]);
  float p_in = sc[21] / fmaxf(tsn, 1.0f);
  float t_in = sc[22] / fmaxf(psn, 1.0f);
  float nsd = empty ? 0.0f : 0.5f * (p_in + t_in);
  float l_nsd = 1.0f - nsd;
  float l_h = fminf(fmaxf(hd95 / 100.0f, 0.0f), 1.0f);
  out[0] = l_brain + l_bone + 2.0f * (l_dice + l_nsd + 0.5f * l_h);
}

// ---------------------------------------------------------------------------
extern "C" void kernel_launch(void* const* d_in, const int* in_sizes, int n_in,
                              void* d_out, int out_size, void* d_ws,
                              size_t ws_size, hipStream_t stream) {
  (void)in_sizes; (void)n_in; (void)out_size; (void)ws_size;
  const float* fused = (const float*)d_in[0];
  const float* mri   = (const float*)d_in[1];
  const float* ct    = (const float*)d_in[2];
  const float* brain = (const float*)d_in[3];
  const float* bone  = (const float*)d_in[4];
  const float* lpred = (const float*)d_in[5];
  const float* lgt   = (const float*)d_in[6];

  char* wsb = (char*)d_ws;
  float* f_pred = (float*)wsb;                           // N floats
  float* f_targ = (float*)(wsb + 4ull * NN);             // N floats
  unsigned char* s_pred = (unsigned char*)(wsb + 8ull * NN);  // N bytes
  unsigned char* s_targ = s_pred + NN;                   // N bytes
  float* partials = (float*)(wsb + 10ull * NN);          // NSUMS*NB floats
  float* scalars = partials + NSUMS * NB;                // 32 floats

  k_surface<<<dim3((NN + 255) / 256), 256, 0, stream>>>(lpred, lgt, f_pred,
                                                        f_targ, s_pred, s_targ);
  k_edt<<<dim3(HW, 2), 96, 0, stream>>>(f_pred, f_targ, 0);
  k_edt<<<dim3(DD * WW, 2), 96, 0, stream>>>(f_pred, f_targ, 1);
  k_edt<<<dim3(DD * HH, 2), 96, 0, stream>>>(f_pred, f_targ, 2);
  k_reduce<<<dim3(NB), 256, 0, stream>>>(fused, mri, ct, brain, bone, lpred,
                                         lgt, f_pred, f_targ, s_pred, s_targ,
                                         partials);
  k_finalize<<<dim3(1), 32, 0, stream>>>(partials, scalars);
  k_select<<<dim3(2), 256, 0, stream>>>(f_pred, f_targ, s_pred, s_targ,
                                        scalars);
  k_final<<<dim3(1), 1, 0, stream>>>(scalars, (float*)d_out);
}